// SkipGramModel_65051574665486
// MI455X (gfx1250) — compile-verified
//
#include <hip/hip_runtime.h>
#include <hip/hip_bf16.h>

typedef __attribute__((ext_vector_type(2))) float v2f;
typedef __attribute__((ext_vector_type(8))) float v8f;

#define DIM    300
#define NCHUNK 75        // DIM / 4 (K=4 per WMMA)
#define KNEG   10
#define BATCH  65536
#define WAVES_PER_BLOCK 8
#define NBLOCKS (BATCH / WAVES_PER_BLOCK)   // 8192

// -log_sigmoid(s) = log(1 + exp(-s));  s is pre-clipped to [-10,10] so exp is safe.
__device__ __forceinline__ float nls(float s) {
    return __logf(1.0f + __expf(-s));
}
__device__ __forceinline__ float clip10(float s) {
    return fminf(fmaxf(s, -10.0f), 10.0f);
}

// One wave handles one sample b.
// A-matrix rows (M=0..15): row 0 = v[pos_v[b]], rows 1..10 = v[neg_v[b][m-1]],
// rows 11..15 = dummy (v row 0, results discarded).
// B-matrix: u[pos_u[b]] chunk broadcast across all 16 columns -> every column
// of D carries the full 16-score vector.
__global__ void __launch_bounds__(256)
skipgram_wmma_kernel(const float* __restrict__ u_weight,
                     const float* __restrict__ v_weight,
                     const int*   __restrict__ pos_u,
                     const int*   __restrict__ pos_v,
                     const int*   __restrict__ neg_v,
                     float*       __restrict__ block_partials)
{
    __shared__ float wave_sums[WAVES_PER_BLOCK];

    const int wave = threadIdx.x >> 5;
    const int lane = threadIdx.x & 31;
    const int half = lane >> 4;      // 0: lanes 0-15, 1: lanes 16-31
    const int m    = lane & 15;      // A-matrix row handled by this lane
    const int b    = blockIdx.x * WAVES_PER_BLOCK + wave;

    // v-row index for A row m (both halves of the wave compute the same value)
    int vidx;
    if (m == 0)          vidx = pos_v[b];
    else if (m <= KNEG)  vidx = neg_v[b * KNEG + (m - 1)];
    else                 vidx = 0;   // dummy pad row (valid memory, result ignored)

    // Per the ISA layouts, both A and B want elements {base+2*half, base+2*half+1}:
    //   A: lane 16*half+m : VGPR0=A[m][2*half], VGPR1=A[m][2*half+1]
    //   B: K = vgpr + 2*half, value independent of column (broadcast)
    const float* vrow = v_weight + (long)vidx * DIM + half * 2;
    const float* urow = u_weight + (long)pos_u[b] * DIM + half * 2;

    v8f acc0 = {};
    v8f acc1 = {};
    int c = 0;
    #pragma unroll 1
    for (; c + 1 < NCHUNK; c += 2) {
        v2f a0 = *(const v2f*)(vrow + (c + 0) * 4);
        v2f b0 = *(const v2f*)(urow + (c + 0) * 4);
        v2f a1 = *(const v2f*)(vrow + (c + 1) * 4);
        v2f b1 = *(const v2f*)(urow + (c + 1) * 4);
        acc0 = __builtin_amdgcn_wmma_f32_16x16x4_f32(false, a0, false, b0,
                                                     (short)0, acc0, false, false);
        acc1 = __builtin_amdgcn_wmma_f32_16x16x4_f32(false, a1, false, b1,
                                                     (short)0, acc1, false, false);
    }
    { // last chunk (NCHUNK=75 is odd)
        v2f a0 = *(const v2f*)(vrow + c * 4);
        v2f b0 = *(const v2f*)(urow + c * 4);
        acc0 = __builtin_amdgcn_wmma_f32_16x16x4_f32(false, a0, false, b0,
                                                     (short)0, acc0, false, false);
    }

    // Scores add linearly, combine the two accumulators.
    v8f acc;
    #pragma unroll
    for (int j = 0; j < 8; ++j) acc[j] = acc0[j] + acc1[j];

    // Every lane of half 0 holds scores 0..7 (VGPR j -> M=j, N=lane),
    // every lane of half 1 holds scores 8..15.
    float part;
    if (half == 0) {
        part = nls(clip10(acc[0]));                       // positive term
        #pragma unroll
        for (int j = 1; j < 8; ++j)                       // neg scores 1..7
            part += nls(-clip10(acc[j]));
    } else {
        part = 0.0f;
        #pragma unroll
        for (int j = 0; j < 3; ++j)                       // neg scores 8..10
            part += nls(-clip10(acc[j]));
        // scores 11..15 are dummy rows -> ignored
    }

    // Combine the two halves of the wave (lane 0 ends with the full per-b loss).
    part += __shfl_xor(part, 16, 32);

    if (lane == 0) wave_sums[wave] = part;
    __syncthreads();
    if (threadIdx.x == 0) {
        float t = 0.0f;
        #pragma unroll
        for (int w = 0; w < WAVES_PER_BLOCK; ++w) t += wave_sums[w];
        block_partials[blockIdx.x] = t;
    }
}

// Deterministic single-block final reduction -> mean.
__global__ void __launch_bounds__(256)
skipgram_reduce_kernel(const float* __restrict__ partials, float* __restrict__ out)
{
    __shared__ float sh[256];
    float t = 0.0f;
    for (int i = threadIdx.x; i < NBLOCKS; i += 256) t += partials[i];
    sh[threadIdx.x] = t;
    __syncthreads();
    for (int s = 128; s > 0; s >>= 1) {
        if (threadIdx.x < s) sh[threadIdx.x] += sh[threadIdx.x + s];
        __syncthreads();
    }
    if (threadIdx.x == 0) out[0] = sh[0] * (1.0f / (float)BATCH);
}

extern "C" void kernel_launch(void* const* d_in, const int* in_sizes, int n_in,
                              void* d_out, int out_size, void* d_ws, size_t ws_size,
                              hipStream_t stream) {
    const float* u_weight = (const float*)d_in[0];   // [100000, 300] f32
    const float* v_weight = (const float*)d_in[1];   // [100000, 300] f32
    const int*   pos_u    = (const int*)d_in[2];     // [65536]
    const int*   pos_v    = (const int*)d_in[3];     // [65536]
    const int*   neg_v    = (const int*)d_in[4];     // [65536, 10]
    float* out            = (float*)d_out;           // scalar mean loss
    float* block_partials = (float*)d_ws;            // NBLOCKS floats (32 KB)

    skipgram_wmma_kernel<<<NBLOCKS, 256, 0, stream>>>(
        u_weight, v_weight, pos_u, pos_v, neg_v, block_partials);
    skipgram_reduce_kernel<<<1, 256, 0, stream>>>(block_partials, out);
}